// GATLayer_26671746908861
// MI455X (gfx1250) — compile-verified
//
#include <hip/hip_runtime.h>
#include <hip/hip_bf16.h>

// GAT layer for MI455X (gfx1250, wave32).
// B=8, N=2000, Fin=128, H=4, Fo=64. NEG_SLOPE=0.2.
// d_out = [ out (B*N*Fo = 1,024,000 f32) | alpha (B*N*N*H = 128,000,000 f32) ]

#define Bc 8
#define Nc 2000
#define FIN 128
#define Hc 4
#define FOc 64
#define HFO 256          // H*Fo
#define ROWS 16000       // B*N
#define NEG_SLOPE 0.2f

typedef __attribute__((ext_vector_type(2))) float v2f;
typedef __attribute__((ext_vector_type(8))) float v8f;

__device__ __forceinline__ float lrelu(float v) {
    return (v > 0.f) ? v : NEG_SLOPE * v;
}

// ---------------------------------------------------------------------------
// Kernel 1: h = x @ W   (16000x128)·(128x256) via V_WMMA_F32_16X16X4_F32.
// One wave per 16x16 output tile; K looped in steps of 4 (32 WMMAs / wave).
// ---------------------------------------------------------------------------
__global__ void gemm_h_kernel(const float* __restrict__ x,
                              const float* __restrict__ W,
                              float* __restrict__ h) {
    const int lane = threadIdx.x & 31;
    const int wave = (blockIdx.x * blockDim.x + threadIdx.x) >> 5; // 0..15999
    const int mtile = wave >> 4;   // 0..999
    const int ntile = wave & 15;   // 0..15
    const int row  = mtile * 16 + (lane & 15);   // A: M = lane&15 for both halves
    const int col  = ntile * 16 + (lane & 15);   // B/C: N = lane&15
    const int ksel = (lane >> 4) * 2;            // lanes 0-15: K 0,1 ; 16-31: K 2,3

    v8f acc = {};
    #pragma unroll 4
    for (int k = 0; k < FIN; k += 4) {
        const int kk = k + ksel;
        v2f a, b;
        a.x = x[row * FIN + kk];
        a.y = x[row * FIN + kk + 1];
        b.x = W[kk * HFO + col];
        b.y = W[(kk + 1) * HFO + col];
        acc = __builtin_amdgcn_wmma_f32_16x16x4_f32(
            false, a, false, b, (short)0, acc, false, false);
    }
    // C layout: VGPR r -> row r (lanes 0-15) / row r+8 (lanes 16-31)
    const int crow0 = mtile * 16 + ((lane >> 4) << 3);
    #pragma unroll
    for (int r = 0; r < 8; ++r)
        h[(crow0 + r) * HFO + col] = acc[r];
}

// ---------------------------------------------------------------------------
// Kernel 2: e_i[row,h] = <h[row,h,:], a_src[h,:]>, e_j likewise with a_dst.
// One wave per row; lanes cover Fo=64 as f=lane and f=lane+32; shfl reduce.
// ---------------------------------------------------------------------------
__global__ void edge_logits_kernel(const float* __restrict__ h,
                                   const float* __restrict__ a,
                                   float* __restrict__ ei,
                                   float* __restrict__ ej) {
    const int lane = threadIdx.x & 31;
    const int row  = (blockIdx.x * blockDim.x + threadIdx.x) >> 5; // 0..15999
    #pragma unroll
    for (int hd = 0; hd < Hc; ++hd) {
        const float as0 = a[hd * 2 * FOc + lane];
        const float as1 = a[hd * 2 * FOc + lane + 32];
        const float ad0 = a[hd * 2 * FOc + FOc + lane];
        const float ad1 = a[hd * 2 * FOc + FOc + lane + 32];
        const float h0  = h[row * HFO + hd * FOc + lane];
        const float h1  = h[row * HFO + hd * FOc + lane + 32];
        float vi = h0 * as0 + h1 * as1;
        float vj = h0 * ad0 + h1 * ad1;
        #pragma unroll
        for (int off = 16; off > 0; off >>= 1) {
            vi += __shfl_xor(vi, off, 32);
            vj += __shfl_xor(vj, off, 32);
        }
        if (lane == 0) {
            ei[row * Hc + hd] = vi;
            ej[row * Hc + hd] = vj;
        }
    }
}

// ---------------------------------------------------------------------------
// Kernel 3: per (b,i,h) online softmax stats over masked j.
// One wave per (b,i); 4 heads kept in registers; shfl-merge at the end.
// ---------------------------------------------------------------------------
__global__ void softmax_stats_kernel(const float* __restrict__ ei,
                                     const float* __restrict__ ej,
                                     const int* __restrict__ adj,
                                     float* __restrict__ m_out,
                                     float* __restrict__ rs_out) {
    const int lane = threadIdx.x & 31;
    const int wid  = (blockIdx.x * blockDim.x + threadIdx.x) >> 5; // b*N + i
    const int i    = wid % Nc;
    const int brow = (wid / Nc) * Nc;

    float eih[Hc];
    #pragma unroll
    for (int hd = 0; hd < Hc; ++hd) eih[hd] = ei[wid * Hc + hd];

    float m[Hc] = {-1e30f, -1e30f, -1e30f, -1e30f};
    float s[Hc] = {0.f, 0.f, 0.f, 0.f};

    for (int j = lane; j < Nc; j += 32) {
        if (adj[i * Nc + j] != 0) {
            #pragma unroll
            for (int hd = 0; hd < Hc; ++hd) {
                float v = lrelu(eih[hd] + ej[(brow + j) * Hc + hd]);
                if (v > m[hd]) {
                    s[hd] = s[hd] * __expf(m[hd] - v) + 1.f;
                    m[hd] = v;
                } else {
                    s[hd] += __expf(v - m[hd]);
                }
            }
        }
    }
    #pragma unroll
    for (int off = 16; off > 0; off >>= 1) {
        #pragma unroll
        for (int hd = 0; hd < Hc; ++hd) {
            float m2 = __shfl_xor(m[hd], off, 32);
            float s2 = __shfl_xor(s[hd], off, 32);
            float nm = fmaxf(m[hd], m2);
            s[hd] = s[hd] * __expf(m[hd] - nm) + s2 * __expf(m2 - nm);
            m[hd] = nm;
        }
    }
    if (lane == 0) {
        #pragma unroll
        for (int hd = 0; hd < Hc; ++hd) {
            m_out[wid * Hc + hd]  = m[hd];
            rs_out[wid * Hc + hd] = (s[hd] > 0.f) ? 1.f / s[hd] : 0.f;
        }
    }
}

// ---------------------------------------------------------------------------
// Kernel 4: stream alpha out (the 512 MB HBM floor). One thread per (b,i,j);
// writes float4 (all 4 heads, innermost dim) fully coalesced.
// ---------------------------------------------------------------------------
__global__ void alpha_kernel(const float* __restrict__ ei,
                             const float* __restrict__ ej,
                             const int* __restrict__ adj,
                             const float* __restrict__ m,
                             const float* __restrict__ rs,
                             float* __restrict__ alpha) {
    const int t = blockIdx.x * blockDim.x + threadIdx.x; // b*N*N + i*N + j
    const int j  = t % Nc;
    const int t2 = t / Nc;
    const int i  = t2 % Nc;
    const int b  = t2 / Nc;
    float4 v = make_float4(0.f, 0.f, 0.f, 0.f);
    if (adj[i * Nc + j] != 0) {
        const int ri = (b * Nc + i) * Hc;
        const int rj = (b * Nc + j) * Hc;
        float* vv = &v.x;
        #pragma unroll
        for (int hd = 0; hd < Hc; ++hd) {
            float e = lrelu(ei[ri + hd] + ej[rj + hd]);
            vv[hd] = __expf(e - m[ri + hd]) * rs[ri + hd];
        }
    }
    reinterpret_cast<float4*>(alpha)[t] = v;
}

// ---------------------------------------------------------------------------
// Kernel 5: out4[b,i,h,f] = sum_j alpha[b,i,j,h] * h[b,j,h,f] via WMMA.
// Alpha fragments regenerated in registers (no HBM re-read of alpha).
// One wave per (b, head, itile); 4 f-tile accumulators (C = 32 VGPRs).
// ---------------------------------------------------------------------------
__global__ void aggregate_kernel(const float* __restrict__ h,
                                 const float* __restrict__ ei,
                                 const float* __restrict__ ej,
                                 const int* __restrict__ adj,
                                 const float* __restrict__ m,
                                 const float* __restrict__ rs,
                                 float* __restrict__ out4) {
    const int lane = threadIdx.x & 31;
    const int wave = (blockIdx.x * blockDim.x + threadIdx.x) >> 5; // 0..3999
    const int itile = wave % 125;
    const int bh    = wave / 125;
    const int head  = bh & 3;
    const int b     = bh >> 2;

    const int il  = itile * 16 + (lane & 15);       // A-matrix row for this lane
    const int ri  = (b * Nc + il) * Hc + head;
    const float ei_l = ei[ri];
    const float m_l  = m[ri];
    const float rs_l = rs[ri];
    const int  ksel  = (lane >> 4) * 2;
    const int  cn    = lane & 15;
    const long hbase = (long)b * Nc * HFO + head * FOc;

    v8f acc0 = {}, acc1 = {}, acc2 = {}, acc3 = {};

    for (int jb = 0; jb < Nc; jb += 4) {
        const int jj = jb + ksel;
        // A fragment: alpha(il, jj), alpha(il, jj+1), built in registers
        const int ad0 = adj[il * Nc + jj];
        const int ad1 = adj[il * Nc + jj + 1];
        const float e0 = lrelu(ei_l + ej[(b * Nc + jj) * Hc + head]);
        const float e1 = lrelu(ei_l + ej[(b * Nc + jj + 1) * Hc + head]);
        v2f a;
        a.x = ad0 ? __expf(e0 - m_l) * rs_l : 0.f;
        a.y = ad1 ? __expf(e1 - m_l) * rs_l : 0.f;
        // B fragments: rows jj / jj+1 of h, 4 f-tiles of 16
        const float* hr0 = h + hbase + (long)jj * HFO;
        const float* hr1 = h + hbase + (long)(jj + 1) * HFO;
        v2f b0, b1, b2, b3;
        b0.x = hr0[cn];      b0.y = hr1[cn];
        b1.x = hr0[16 + cn]; b1.y = hr1[16 + cn];
        b2.x = hr0[32 + cn]; b2.y = hr1[32 + cn];
        b3.x = hr0[48 + cn]; b3.y = hr1[48 + cn];
        acc0 = __builtin_amdgcn_wmma_f32_16x16x4_f32(false, a, false, b0, (short)0, acc0, false, false);
        acc1 = __builtin_amdgcn_wmma_f32_16x16x4_f32(false, a, false, b1, (short)0, acc1, false, false);
        acc2 = __builtin_amdgcn_wmma_f32_16x16x4_f32(false, a, false, b2, (short)0, acc2, false, false);
        acc3 = __builtin_amdgcn_wmma_f32_16x16x4_f32(false, a, false, b3, (short)0, acc3, false, false);
    }

    const int crow0 = itile * 16 + ((lane >> 4) << 3);
    #pragma unroll
    for (int r = 0; r < 8; ++r) {
        const long o = ((long)b * Nc + crow0 + r) * HFO + head * FOc + cn;
        out4[o]      = acc0[r];
        out4[o + 16] = acc1[r];
        out4[o + 32] = acc2[r];
        out4[o + 48] = acc3[r];
    }
}

// ---------------------------------------------------------------------------
// Kernel 6: out[b,i,f] = mean over heads of out4[b,i,h,f].
// ---------------------------------------------------------------------------
__global__ void mean_kernel(const float* __restrict__ out4,
                            float* __restrict__ out) {
    const int t  = blockIdx.x * blockDim.x + threadIdx.x; // < 1,024,000
    const int f  = t & 63;
    const int bi = t >> 6;
    const float* p = out4 + (long)bi * HFO;
    out[t] = 0.25f * (p[f] + p[FOc + f] + p[2 * FOc + f] + p[3 * FOc + f]);
}

// ---------------------------------------------------------------------------
extern "C" void kernel_launch(void* const* d_in, const int* in_sizes, int n_in,
                              void* d_out, int out_size, void* d_ws, size_t ws_size,
                              hipStream_t stream) {
    const float* x   = (const float*)d_in[0];  // (B,N,Fin)
    const int*   adj = (const int*)  d_in[1];  // (N,N)
    const float* W   = (const float*)d_in[2];  // (Fin,H*Fo)
    const float* a   = (const float*)d_in[3];  // (H,2*Fo)

    float* out   = (float*)d_out;              // (B,N,Fo)   = 1,024,000
    float* alpha = out + (long)Bc * Nc * FOc;  // (B,N,N,H)  = 128,000,000 (16B aligned)

    // Workspace layout (floats): h | ei | ej | m | rs | out4  (~33.8 MB)
    float* ws   = (float*)d_ws;
    float* h    = ws;                           // 4,096,000
    float* ei   = h + (long)ROWS * HFO;         // 64,000
    float* ej   = ei + ROWS * Hc;               // 64,000
    float* mbuf = ej + ROWS * Hc;               // 64,000
    float* rsb  = mbuf + ROWS * Hc;             // 64,000
    float* out4 = rsb + ROWS * Hc;              // 4,096,000

    // 16000 tile-waves (1000 Mtiles x 16 Ntiles) -> 2000 blocks of 8 waves
    gemm_h_kernel<<<2000, 256, 0, stream>>>(x, W, h);
    // 16000 row-waves
    edge_logits_kernel<<<2000, 256, 0, stream>>>(h, a, ei, ej);
    // 16000 (b,i)-waves
    softmax_stats_kernel<<<2000, 256, 0, stream>>>(ei, ej, adj, mbuf, rsb);
    // 32,000,000 threads, float4 coalesced alpha stream
    alpha_kernel<<<125000, 256, 0, stream>>>(ei, ej, adj, mbuf, rsb, alpha);
    // 4000 (b,head,itile)-waves, WMMA aggregation
    aggregate_kernel<<<500, 256, 0, stream>>>(h, ei, ej, adj, mbuf, rsb, out4);
    // head-mean
    mean_kernel<<<4000, 256, 0, stream>>>(out4, out);
}